// DensetoConv_68590627717564
// MI455X (gfx1250) — compile-verified
//
#include <hip/hip_runtime.h>

// GEMV out[o] = sum_k x[k]*W[o,k] + b[o], M=1, N=1000, K=147456, fp32.
// HBM-bound: 590 MB of W @ 23.3 TB/s ~= 25us floor. Use V_WMMA_F32_16X16X4_F32
// (native fp32 -> no conversion traffic) as the accumulate engine:
//   A = 16x4 tile of W (16 classes, 4 K)  [lane m: K=0,1 | lane m+16: K=2,3]
//   B = 4x16 broadcast of x (B[k][n] = x[k]) -> every D column = the 16 dots.
// K split 64 ways across waves, f32 global atomics finish the reduction.

typedef __attribute__((ext_vector_type(2))) float v2f;
typedef __attribute__((ext_vector_type(8))) float v8f;

#define N_IN       147456
#define N_CLASSES  1000
#define N_TILES    63            // ceil(1000/16)
#define K_PARTS    64
#define K_PER_PART (N_IN / K_PARTS)   // 2304, multiple of 32
#define UNROLL     8                  // 8 WMMAs -> K advances 32 per iter

__global__ void __launch_bounds__(256)
dense_gemv_init(const float* __restrict__ b, float* __restrict__ out, int n) {
    int i = blockIdx.x * blockDim.x + threadIdx.x;
    if (i < n) out[i] = b[i];
}

__global__ void __launch_bounds__(256)
dense_gemv_wmma(const float* __restrict__ x,
                const float* __restrict__ W,
                float* __restrict__ out) {
    const int lane = threadIdx.x & 31;
    const int wave = threadIdx.x >> 5;
    const int tile  = blockIdx.x >> 3;                    // 0..62 (class tile)
    const int kpart = ((blockIdx.x & 7) << 3) | wave;     // 0..63 (K partition)

    const int half = lane >> 4;      // 0: K=0,1   1: K=2,3 (per A 16x4 layout)
    const int m    = lane & 15;      // row (class) within tile

    int row = tile * 16 + m;
    // Tail-tile guard via value masking (WMMA needs EXEC all-ones; no branches).
    const float mask = (row < N_CLASSES) ? 1.0f : 0.0f;
    if (row >= N_CLASSES) row = N_CLASSES - 1;

    const float* __restrict__ wrow = W + (size_t)row * N_IN + 2 * half;
    const float* __restrict__ xptr = x + 2 * half;

    const int k0 = kpart * K_PER_PART;

    v8f acc = {0.f, 0.f, 0.f, 0.f, 0.f, 0.f, 0.f, 0.f};

    for (int k = k0; k < k0 + K_PER_PART; k += 4 * UNROLL) {
#pragma unroll
        for (int u = 0; u < UNROLL; ++u) {
            const int kk = k + 4 * u;
            v2f a = *(const v2f*)(wrow + kk);   // W[row, kk+2h .. kk+2h+1]
            a.x *= mask;
            a.y *= mask;
            v2f bb = *(const v2f*)(xptr + kk);  // x[kk+2h .. kk+2h+1] (bcast over N)
            // (neg_a, A, neg_b, B, c_mod, C, reuse_a, reuse_b)
            acc = __builtin_amdgcn_wmma_f32_16x16x4_f32(
                false, a, false, bb, (short)0, acc, false, false);
        }
    }

    // Every column N of D is identical. D layout: VGPR v holds
    // (M=v, N=lane) for lanes 0-15 and (M=v+8, N=lane-16) for lanes 16-31.
    // Lane with N==0 in each half commits 8 rows via f32 atomics.
    if ((lane & 15) == 0) {
        const int base = tile * 16 + half * 8;
#pragma unroll
        for (int v = 0; v < 8; ++v) {
            const int c = base + v;
            if (c < N_CLASSES) atomicAdd(&out[c], acc[v]);
        }
    }
}

extern "C" void kernel_launch(void* const* d_in, const int* in_sizes, int n_in,
                              void* d_out, int out_size, void* d_ws, size_t ws_size,
                              hipStream_t stream) {
    const float* x = (const float*)d_in[0];   // [1, 147456]
    const float* W = (const float*)d_in[1];   // [1000, 147456]
    const float* b = (const float*)d_in[2];   // [1000] (zeros, but honor it)
    float* out = (float*)d_out;               // [1000] (1 x 1000)

    dense_gemv_init<<<(N_CLASSES + 255) / 256, 256, 0, stream>>>(b, out, N_CLASSES);

    // 63 class tiles x 8 blocks; each block = 8 waves = 8 K-partitions.
    dense_gemv_wmma<<<N_TILES * 8, 256, 0, stream>>>(x, W, out);
}